// GEO_13872744366621
// MI455X (gfx1250) — compile-verified
//
#include <hip/hip_runtime.h>

typedef float v2f __attribute__((ext_vector_type(2)));
typedef float v8f __attribute__((ext_vector_type(8)));

#if defined(__HIP_DEVICE_COMPILE__)
#if !__has_builtin(__builtin_amdgcn_wmma_f32_16x16x4_f32)
#error "missing __builtin_amdgcn_wmma_f32_16x16x4_f32"
#endif
#endif

#define C_DIM 1024
#define H_DIM 16
#define D_DIM 64
#define QN 2048
#define KN 2048

__device__ __forceinline__ v8f wmma_f32(v2f a, v2f b, v8f c) {
  // 8 args: (neg_a, A, neg_b, B, c_mod, C, reuse_a, reuse_b)
  return __builtin_amdgcn_wmma_f32_16x16x4_f32(false, a, false, b, (short)0, c,
                                               false, false);
}

__device__ __forceinline__ void store_tile(float* __restrict__ out,
                                           const float* __restrict__ bias,
                                           int m0, int ncol, int lr, int lh,
                                           const v8f& acc) {
  const float bn = bias[ncol + lr];
#pragma unroll
  for (int j = 0; j < 8; ++j) {
    out[(size_t)(m0 + j + 8 * lh) * C_DIM + ncol + lr] = acc[j] + bn;
  }
}

// out[M=2048, N=1024] = X[2048,1024] @ W[1024,1024]^T + bias
// One 16x64 strip per wave: A fragment reused across 4 B tiles (4 WMMA / 5 loads).
__global__ __launch_bounds__(256) void proj_gemm(const float* __restrict__ X,
                                                 const float* __restrict__ W,
                                                 const float* __restrict__ bias,
                                                 float* __restrict__ out) {
  const int lane = threadIdx.x & 31;
  const int wave = threadIdx.x >> 5;
  const int lr = lane & 15;  // row/col within tile
  const int lh = lane >> 4;  // lane half selects K pair
  const int m0 = blockIdx.x * 16;
  const int n0 = (blockIdx.y * 8 + wave) * 64;

  const float* aptr = X + (size_t)(m0 + lr) * C_DIM + 2 * lh;
  const float* b0 = W + (size_t)(n0 + 0 + lr) * C_DIM + 2 * lh;
  const float* b1 = W + (size_t)(n0 + 16 + lr) * C_DIM + 2 * lh;
  const float* b2 = W + (size_t)(n0 + 32 + lr) * C_DIM + 2 * lh;
  const float* b3 = W + (size_t)(n0 + 48 + lr) * C_DIM + 2 * lh;

  v8f acc0 = {}, acc1 = {}, acc2 = {}, acc3 = {};
#pragma unroll 4
  for (int k = 0; k < C_DIM; k += 4) {
    v2f a = *(const v2f*)(aptr + k);
    acc0 = wmma_f32(a, *(const v2f*)(b0 + k), acc0);
    acc1 = wmma_f32(a, *(const v2f*)(b1 + k), acc1);
    acc2 = wmma_f32(a, *(const v2f*)(b2 + k), acc2);
    acc3 = wmma_f32(a, *(const v2f*)(b3 + k), acc3);
  }

  store_tile(out, bias, m0, n0 + 0, lr, lh, acc0);
  store_tile(out, bias, m0, n0 + 16, lr, lh, acc1);
  store_tile(out, bias, m0, n0 + 32, lr, lh, acc2);
  store_tile(out, bias, m0, n0 + 48, lr, lh, acc3);
}

// Per block: one head, 16 query rows, full K=2048 attention.
// 4 waves split key tiles; softmax via max-recompute (2-pass), O reduced in LDS.
// Q A-fragments are loop-invariant: loaded once, reused across both passes.
__global__ __launch_bounds__(128) void attn_kernel(
    const float* __restrict__ qp, const float* __restrict__ kp,
    const float* __restrict__ vp, const float* __restrict__ query,
    float* __restrict__ out) {
  __shared__ float Pbuf[4][16 * 18];   // per-wave P tile, padded stride 18
  __shared__ float gmax[4][16];
  __shared__ float gsum[4][16];
  __shared__ float Obuf[4][16 * 65];   // per-wave O partials, padded stride 65

  const int lane = threadIdx.x & 31;
  const int wave = threadIdx.x >> 5;
  const int lr = lane & 15;
  const int lh = lane >> 4;
  const int m0 = blockIdx.x * 16;
  const int hc = blockIdx.y * D_DIM;
  const float scale = 0.125f;  // 1/sqrt(64)

  // hoist the Q tile A-fragments (invariant over key tiles and both passes)
  const float* aBase = qp + (size_t)(m0 + lr) * C_DIM + hc + 2 * lh;
  v2f aFrag[16];
#pragma unroll
  for (int k4 = 0; k4 < 16; ++k4) aFrag[k4] = *(const v2f*)(aBase + 4 * k4);

  // ---- Pass 1: row maxima of raw logits ----
  float rmax[8];
#pragma unroll
  for (int j = 0; j < 8; ++j) rmax[j] = -3.0e38f;

  for (int kt = wave; kt < KN / 16; kt += 4) {
    const int n0 = kt * 16;
    const float* bBase = kp + (size_t)(n0 + lr) * C_DIM + hc + 2 * lh;
    v8f acc = {};
#pragma unroll
    for (int k4 = 0; k4 < 16; ++k4) {
      v2f b = *(const v2f*)(bBase + 4 * k4);
      acc = wmma_f32(aFrag[k4], b, acc);
    }
#pragma unroll
    for (int j = 0; j < 8; ++j) rmax[j] = fmaxf(rmax[j], acc[j]);
  }
  // reduce across the 16 lanes that hold the same row (stay within half-wave)
#pragma unroll
  for (int off = 1; off <= 8; off <<= 1) {
#pragma unroll
    for (int j = 0; j < 8; ++j)
      rmax[j] = fmaxf(rmax[j], __shfl_xor(rmax[j], off));
  }
  if (lr == 0) {  // lanes 0 and 16 publish 8 rows each
#pragma unroll
    for (int j = 0; j < 8; ++j) gmax[wave][8 * lh + j] = rmax[j];
  }
  __syncthreads();
#pragma unroll
  for (int j = 0; j < 8; ++j) {
    float m = gmax[0][8 * lh + j];
    m = fmaxf(m, gmax[1][8 * lh + j]);
    m = fmaxf(m, gmax[2][8 * lh + j]);
    m = fmaxf(m, gmax[3][8 * lh + j]);
    rmax[j] = m;
  }

  // ---- Pass 2: recompute S, exp, and P@V ----
  float rsum[8];
#pragma unroll
  for (int j = 0; j < 8; ++j) rsum[j] = 0.0f;
  v8f o0 = {}, o1 = {}, o2 = {}, o3 = {};
  float* pb = &Pbuf[wave][0];

  for (int kt = wave; kt < KN / 16; kt += 4) {
    const int n0 = kt * 16;
    const float* bBase = kp + (size_t)(n0 + lr) * C_DIM + hc + 2 * lh;
    v8f acc = {};
#pragma unroll
    for (int k4 = 0; k4 < 16; ++k4) {
      v2f b = *(const v2f*)(bBase + 4 * k4);
      acc = wmma_f32(aFrag[k4], b, acc);
    }
    // exp + running row sums; write P tile to per-wave LDS (D-layout)
#pragma unroll
    for (int j = 0; j < 8; ++j) {
      float e = __expf((acc[j] - rmax[j]) * scale);
      rsum[j] += e;
      pb[(j + 8 * lh) * 18 + lr] = e;
    }
    // per-wave in-order LDS: wait stores, then read back in A-fragment layout
    asm volatile("s_wait_dscnt 0" ::: "memory");
#pragma unroll
    for (int kb = 0; kb < 16; kb += 4) {
      v2f pa = *(const v2f*)(pb + lr * 18 + kb + 2 * lh);
      const float* vr = vp + (size_t)(n0 + kb + 2 * lh) * C_DIM + hc + lr;
      v2f b;
      b.x = vr[0];
      b.y = vr[C_DIM];
      o0 = wmma_f32(pa, b, o0);
      b.x = vr[16];
      b.y = vr[C_DIM + 16];
      o1 = wmma_f32(pa, b, o1);
      b.x = vr[32];
      b.y = vr[C_DIM + 32];
      o2 = wmma_f32(pa, b, o2);
      b.x = vr[48];
      b.y = vr[C_DIM + 48];
      o3 = wmma_f32(pa, b, o3);
    }
    asm volatile("s_wait_dscnt 0" ::: "memory");  // protect Pbuf WAR next iter
  }

  // reduce row sums within half-wave, publish per-wave
#pragma unroll
  for (int off = 1; off <= 8; off <<= 1) {
#pragma unroll
    for (int j = 0; j < 8; ++j) rsum[j] += __shfl_xor(rsum[j], off);
  }
  if (lr == 0) {
#pragma unroll
    for (int j = 0; j < 8; ++j) gsum[wave][8 * lh + j] = rsum[j];
  }
  // publish per-wave O partials
  float* ob = &Obuf[wave][0];
#pragma unroll
  for (int j = 0; j < 8; ++j) {
    const int row = j + 8 * lh;
    ob[row * 65 + lr + 0] = o0[j];
    ob[row * 65 + lr + 16] = o1[j];
    ob[row * 65 + lr + 32] = o2[j];
    ob[row * 65 + lr + 48] = o3[j];
  }
  __syncthreads();

  // ---- Final: out = query + (sum_w O_w) / (sum_w rsum_w); bv folded into vp
  for (int idx = threadIdx.x; idx < 16 * 64; idx += 128) {
    const int row = idx >> 6;
    const int col = idx & 63;
    float v = Obuf[0][row * 65 + col] + Obuf[1][row * 65 + col] +
              Obuf[2][row * 65 + col] + Obuf[3][row * 65 + col];
    float rs = gsum[0][row] + gsum[1][row] + gsum[2][row] + gsum[3][row];
    const size_t g = (size_t)(m0 + row) * C_DIM + hc + col;
    out[g] = query[g] + v / rs;
  }
}

extern "C" void kernel_launch(void* const* d_in, const int* in_sizes, int n_in,
                              void* d_out, int out_size, void* d_ws,
                              size_t ws_size, hipStream_t stream) {
  (void)in_sizes;
  (void)n_in;
  (void)out_size;
  (void)ws_size;
  const float* query = (const float*)d_in[0];
  const float* key = (const float*)d_in[1];
  const float* Wq = (const float*)d_in[2];
  const float* bq = (const float*)d_in[3];
  const float* Wk = (const float*)d_in[4];
  const float* bk = (const float*)d_in[5];
  const float* Wv = (const float*)d_in[6];
  const float* bv = (const float*)d_in[7];
  float* out = (float*)d_out;

  float* qp = (float*)d_ws;                    // [2048,1024]
  float* kp = qp + (size_t)QN * C_DIM;         // [2048,1024]
  float* vp = kp + (size_t)KN * C_DIM;         // [2048,1024]

  dim3 pgrid(QN / 16, C_DIM / (64 * 8));
  proj_gemm<<<pgrid, 256, 0, stream>>>(query, Wq, bq, qp);
  proj_gemm<<<pgrid, 256, 0, stream>>>(key, Wk, bk, kp);
  // bv folded here: attn rows sum to 1, so attn@(V+bv) == attn@V + bv
  proj_gemm<<<pgrid, 256, 0, stream>>>(key, Wv, bv, vp);

  dim3 agrid(QN / 16, H_DIM);
  attn_kernel<<<agrid, 128, 0, stream>>>(qp, kp, vp, query, out);
}